// GNN_mol_bond_aggregate_13322988552789
// MI455X (gfx1250) — compile-verified
//
#include <hip/hip_runtime.h>

#define B_SZ 4096
#define L_SZ 200
#define D_SZ 128
#define LN_EPS 1e-5f

typedef __attribute__((ext_vector_type(2))) float f32x2;
typedef __attribute__((ext_vector_type(8))) float f32x8;
typedef unsigned int u32x4 __attribute__((ext_vector_type(4)));
typedef int i32x4 __attribute__((ext_vector_type(4)));
typedef int i32x8 __attribute__((ext_vector_type(8)));

// workspace layout (floats)
#define WS_U 0
#define WS_V 128
#define WS_C 256
#define WS_G 512
#define WS_S (512 + B_SZ * D_SZ)

__device__ __forceinline__ float waveReduceSum(float v) {
#pragma unroll
  for (int off = 16; off > 0; off >>= 1) v += __shfl_xor(v, off, 32);
  return v;
}
__device__ __forceinline__ float waveReduceMax(float v) {
#pragma unroll
  for (int off = 16; off > 0; off >>= 1) v = fmaxf(v, __shfl_xor(v, off, 32));
  return v;
}

// ---------------------------------------------------------------------------
// TDM: issue one 2D tensor tile load (rows x 128 fp32, contiguous) -> LDS.
// D# built per CDNA5 ISA ch.8: group0 {count, lds_addr, global_addr, type=2},
// group1 {data_size=4B, tensor_dim0=128, tensor_dim1=rows, tile=128 x rows,
// dim0_stride=128}. Groups 2/3 + trailing group zero (2D tile, tile_dim2=0).
// 6-arg builtin form (clang-23 / therock toolchain).
// ---------------------------------------------------------------------------
__device__ __forceinline__ void tdm_load_rows(uint32_t lds_addr, const float* gptr,
                                              uint32_t rows) {
  const uint64_t ga = (uint64_t)(uintptr_t)gptr;
  u32x4 g0;
  g0[0] = 1u;                                    // count=1, user descriptor
  g0[1] = lds_addr;                              // LDS byte address
  g0[2] = (uint32_t)(ga & 0xFFFFFFFFu);          // global_addr[31:0]
  g0[3] = (uint32_t)((ga >> 32) & 0x01FFFFFFu)   // global_addr[56:32]
          | (2u << 30);                          // type = 2 ("image")
  i32x8 g1;
  g1[0] = (int)(2u << 16);                       // data_size = 2 -> 4 bytes
  g1[1] = (int)(128u << 16);                     // tensor_dim0 = 128 (bits 63:48)
  g1[2] = (int)(rows << 16);                     // tensor_dim1 = rows (bits 95:80)
  g1[3] = (int)(128u << 16);                     // tile_dim0 = 128 (bits 127:112)
  g1[4] = (int)rows;                             // tile_dim1 = rows (bits 143:128)
  g1[5] = 128;                                   // tensor_dim0_stride[31:0] = 128
  g1[6] = (int)(((128u * rows) & 0xFFFFu) << 16);// tensor_dim1_stride[15:0]
  g1[7] = (int)((128u * rows) >> 16);            // tensor_dim1_stride[47:16]
  const i32x4 z4 = {0, 0, 0, 0};
  const i32x8 z8 = {0, 0, 0, 0, 0, 0, 0, 0};
  __builtin_amdgcn_tensor_load_to_lds(g0, g1, z4, z4, z8, 0);
}

// ---------------------------------------------------------------------------
// Kernel 0: u[d] = sum_e wy[e]*W_nb[e,d]; v[d] = sum_e wx[e]*W_self[e,d];
//           c = wx.b_self + wy.b_nb + b_align
// ---------------------------------------------------------------------------
__global__ void prep_kernel(const float* __restrict__ W_self,
                            const float* __restrict__ b_self,
                            const float* __restrict__ W_nb,
                            const float* __restrict__ b_nb,
                            const float* __restrict__ w_align,
                            const float* __restrict__ b_align,
                            float* __restrict__ ws) {
  const int t = threadIdx.x;
  if (t < 128) {
    float acc = 0.f;
    for (int e = 0; e < 128; ++e) acc = fmaf(w_align[128 + e], W_nb[e * 128 + t], acc);
    ws[WS_U + t] = acc;
  } else {
    const int d = t - 128;
    float acc = 0.f;
    for (int e = 0; e < 128; ++e) acc = fmaf(w_align[e], W_self[e * 128 + d], acc);
    ws[WS_V + d] = acc;
  }
  if (t == 0) {
    float c = b_align[0];
    for (int e = 0; e < 128; ++e) c = fmaf(w_align[e], b_self[e], c);
    for (int e = 0; e < 128; ++e) c = fmaf(w_align[128 + e], b_nb[e], c);
    ws[WS_C] = c;
  }
}

// ---------------------------------------------------------------------------
// Kernel 1: per-b fused scores -> softmax -> masked attn -> g[b,:], S[b].
// bond_feature[b] (200x128 fp32, 100 KB) is staged HBM -> LDS by two TDM
// tensor loads issued by wave 0; score pass on tile 0 overlaps tile 1 DMA.
// Single HBM pass over the 419 MB tensor.
// ---------------------------------------------------------------------------
__global__ __launch_bounds__(256, 4) void attn_kernel(
    const float* __restrict__ mbf,    // (B,D)
    const float* __restrict__ bond,   // (B,L,D)
    const float* __restrict__ amask,  // (B,L,1)
    const float* __restrict__ smask,  // (B,L,1)
    float* __restrict__ ws) {
  extern __shared__ __align__(16) float s_bond[];  // L_SZ*D_SZ floats (102400 B)
  __shared__ float s_score[256];                   // L_SZ=200 used
  __shared__ float s_red[8];
  __shared__ float s_b0;
  __shared__ float s_g[8 * 128];

  const int b = blockIdx.x;
  const int tid = threadIdx.x;
  const int lane = tid & 31;
  const int wave = tid >> 5;
  const float* bondb = bond + (size_t)b * (L_SZ * D_SZ);

  // wave 0: launch both TDM tiles up front (100 rows each)
  if (wave == 0) {
    const uint32_t lds0 = (uint32_t)(uintptr_t)s_bond;
    tdm_load_rows(lds0, bondb, 100u);
    tdm_load_rows(lds0 + 100u * 512u, bondb + 100 * D_SZ, 100u);
  }

  const float4 u4 = *reinterpret_cast<const float4*>(ws + WS_U + lane * 4);

  // sbase = mbf[b].v + c   (wave 0, overlapped with DMA)
  if (wave == 0) {
    const float4 m4 = *reinterpret_cast<const float4*>(mbf + (size_t)b * D_SZ + lane * 4);
    const float4 v4 = *reinterpret_cast<const float4*>(ws + WS_V + lane * 4);
    float p = m4.x * v4.x + m4.y * v4.y + m4.z * v4.z + m4.w * v4.w;
    p = waveReduceSum(p);
    if (lane == 0) s_b0 = p + ws[WS_C];
  }

  // tile 0 ready?
  if (wave == 0) __builtin_amdgcn_s_wait_tensorcnt(1);
  __syncthreads();
  const float sbase = s_b0;

  // pass 1a: scores for rows [0,100) from LDS
  for (int l = wave; l < 100; l += 8) {
    const float4 f = *reinterpret_cast<const float4*>(&s_bond[l * D_SZ + lane * 4]);
    float p = f.x * u4.x + f.y * u4.y + f.z * u4.z + f.w * u4.w;
    p = waveReduceSum(p);
    if (lane == 0) {
      float s = sbase + p;
      s = (s > 0.f) ? s : 0.01f * s;  // leaky_relu, slope 0.01
      s += smask[(size_t)b * L_SZ + l];
      s_score[l] = s;
    }
  }

  // tile 1 ready?
  if (wave == 0) __builtin_amdgcn_s_wait_tensorcnt(0);
  __syncthreads();

  // pass 1b: scores for rows [100,200)
  for (int l = 100 + wave; l < L_SZ; l += 8) {
    const float4 f = *reinterpret_cast<const float4*>(&s_bond[l * D_SZ + lane * 4]);
    float p = f.x * u4.x + f.y * u4.y + f.z * u4.z + f.w * u4.w;
    p = waveReduceSum(p);
    if (lane == 0) {
      float s = sbase + p;
      s = (s > 0.f) ? s : 0.01f * s;
      s += smask[(size_t)b * L_SZ + l];
      s_score[l] = s;
    }
  }
  __syncthreads();

  // block max
  float val = (tid < L_SZ) ? s_score[tid] : -3.0e38f;
  float m = waveReduceMax(val);
  if (lane == 0) s_red[wave] = m;
  __syncthreads();
  if (tid == 0) {
    float mm = s_red[0];
    for (int w = 1; w < 8; ++w) mm = fmaxf(mm, s_red[w]);
    s_b0 = mm;
  }
  __syncthreads();
  const float smax = s_b0;

  // exp + block sum
  float e = 0.f;
  if (tid < L_SZ) {
    e = expf(s_score[tid] - smax);
    s_score[tid] = e;
  }
  float se = waveReduceSum(e);
  if (lane == 0) s_red[wave] = se;
  __syncthreads();
  if (tid == 0) {
    float ss = 0.f;
    for (int w = 0; w < 8; ++w) ss += s_red[w];
    s_b0 = ss;
  }
  __syncthreads();
  const float denom = s_b0;

  // attn = softmax * attend_mask ; S = sum(attn)
  float a = 0.f;
  if (tid < L_SZ) {
    a = (s_score[tid] / denom) * amask[(size_t)b * L_SZ + tid];
    s_score[tid] = a;
  }
  float sa = waveReduceSum(a);
  if (lane == 0) s_red[wave] = sa;
  __syncthreads();
  if (tid == 0) {
    float ss = 0.f;
    for (int w = 0; w < 8; ++w) ss += s_red[w];
    ws[WS_S + b] = ss;
  }
  __syncthreads();

  // pass 2: g[d] = sum_l attn[l] * bond[b,l,d]   (entirely from LDS)
  float4 acc = {0.f, 0.f, 0.f, 0.f};
  for (int l = wave; l < L_SZ; l += 8) {
    const float w = s_score[l];
    const float4 f = *reinterpret_cast<const float4*>(&s_bond[l * D_SZ + lane * 4]);
    acc.x = fmaf(w, f.x, acc.x);
    acc.y = fmaf(w, f.y, acc.y);
    acc.z = fmaf(w, f.z, acc.z);
    acc.w = fmaf(w, f.w, acc.w);
  }
  *reinterpret_cast<float4*>(&s_g[wave * 128 + lane * 4]) = acc;
  __syncthreads();
  if (tid < 128) {
    float g = 0.f;
#pragma unroll
    for (int w = 0; w < 8; ++w) g += s_g[w * 128 + tid];
    ws[WS_G + (size_t)b * D_SZ + tid] = g;
  }
}

// ---------------------------------------------------------------------------
// Kernel 2: ctx = g @ W_nb^T + S*b_nb  via V_WMMA_F32_16X16X4_F32, then LN.
// 16 b-rows per block; 8 waves, each wave owns one 16-wide N tile.
// A frag (16x4): lane m(0..15) holds row m; VGPR0=K(0|2), VGPR1=K(1|3)
// B frag (4x16): lane n(0..15) holds col n; same K split. B[k,n]=W_nb[n,k].
// ---------------------------------------------------------------------------
__global__ __launch_bounds__(256, 4) void ctx_ln_kernel(
    const float* __restrict__ W_nb,   // (D,D) row-major (e,d)
    const float* __restrict__ b_nb,   // (D)
    const float* __restrict__ gamma,  // (D)
    const float* __restrict__ beta,   // (D)
    const float* __restrict__ ws,
    float* __restrict__ out) {
  const int tid = threadIdx.x;
  const int lane = tid & 31;
  const int wave = tid >> 5;
  const int b0 = blockIdx.x * 16;
  const int n0 = wave * 16;

  const float* g = ws + WS_G;
  const int row = b0 + (lane & 15);  // A-matrix M row for this lane
  const int col = n0 + (lane & 15);  // B-matrix N col for this lane
  const int ksel = (lane >> 4) * 2;  // half-wave K split

  f32x8 c = {0.f, 0.f, 0.f, 0.f, 0.f, 0.f, 0.f, 0.f};
#pragma unroll 8
  for (int k0 = 0; k0 < D_SZ; k0 += 4) {
    const int kk = k0 + ksel;
    const float2 av = *reinterpret_cast<const float2*>(g + (size_t)row * D_SZ + kk);
    const float2 bv = *reinterpret_cast<const float2*>(W_nb + (size_t)col * D_SZ + kk);
    f32x2 afrag, bfrag;
    afrag.x = av.x; afrag.y = av.y;
    bfrag.x = bv.x; bfrag.y = bv.y;
    c = __builtin_amdgcn_wmma_f32_16x16x4_f32(false, afrag, false, bfrag,
                                              (short)0, c, false, false);
  }

  __shared__ float s_ctx[16 * 128];
  __shared__ float s_mu[16];
  __shared__ float s_rs[16];

  const int row_hi = (lane >> 4) * 8;  // C/D layout: lanes 16-31 hold M+8
  const float bnb = b_nb[col];
#pragma unroll
  for (int r = 0; r < 8; ++r) {
    const int rl = r + row_hi;
    const float Sv = ws[WS_S + b0 + rl];
    s_ctx[rl * 128 + (n0 + (lane & 15))] = c[r] + Sv * bnb;
  }
  __syncthreads();

  // LayerNorm stats: one thread per row
  if (tid < 16) {
    float mu = 0.f;
    for (int d = 0; d < 128; ++d) mu += s_ctx[tid * 128 + d];
    mu *= (1.0f / 128.f);
    float var = 0.f;
    for (int d = 0; d < 128; ++d) {
      const float t = s_ctx[tid * 128 + d] - mu;
      var = fmaf(t, t, var);
    }
    var *= (1.0f / 128.f);
    s_mu[tid] = mu;
    s_rs[tid] = rsqrtf(var + LN_EPS);
  }
  __syncthreads();

  const int orow = tid >> 4;
  const int d0 = (tid & 15) * 8;
  const float mu = s_mu[orow];
  const float rs = s_rs[orow];
#pragma unroll
  for (int i = 0; i < 8; ++i) {
    const int d = d0 + i;
    out[(size_t)(b0 + orow) * D_SZ + d] =
        (s_ctx[orow * 128 + d] - mu) * rs * gamma[d] + beta[d];
  }
}

// ---------------------------------------------------------------------------
extern "C" void kernel_launch(void* const* d_in, const int* in_sizes, int n_in,
                              void* d_out, int out_size, void* d_ws, size_t ws_size,
                              hipStream_t stream) {
  (void)in_sizes; (void)n_in; (void)out_size; (void)ws_size;
  const float* mbf     = (const float*)d_in[0];
  const float* bond    = (const float*)d_in[1];
  const float* amask   = (const float*)d_in[2];
  const float* smask   = (const float*)d_in[3];
  const float* W_self  = (const float*)d_in[4];
  const float* b_self  = (const float*)d_in[5];
  const float* W_nb    = (const float*)d_in[6];
  const float* b_nb    = (const float*)d_in[7];
  const float* w_align = (const float*)d_in[8];
  const float* b_align = (const float*)d_in[9];
  const float* gamma   = (const float*)d_in[10];
  const float* beta    = (const float*)d_in[11];
  float* out = (float*)d_out;
  float* ws  = (float*)d_ws;

  const int dynLds = L_SZ * D_SZ * (int)sizeof(float);  // 102400 B
  (void)hipFuncSetAttribute((const void*)attn_kernel,
                            hipFuncAttributeMaxDynamicSharedMemorySize, dynLds);

  prep_kernel<<<1, 256, 0, stream>>>(W_self, b_self, W_nb, b_nb, w_align, b_align, ws);
  attn_kernel<<<B_SZ, 256, dynLds, stream>>>(mbf, bond, amask, smask, ws);
  ctx_ln_kernel<<<B_SZ / 16, 256, 0, stream>>>(W_nb, b_nb, gamma, beta, ws, out);
}